// DistributionalVariancePenalization_87359634800727
// MI455X (gfx1250) — compile-verified
//
#include <hip/hip_runtime.h>

// ---------------------------------------------------------------------------
// Distributional variance penalization, fully on-device.
// Persistent cooperative kernel: stats pass -> on-device bisection loop
// (each iteration = one L2-resident streaming pass) -> final weighted dot.
// Wave reduction uses v_wmma_f32_16x16x4_f32 (ones-B => row sums).
// ---------------------------------------------------------------------------

#define NBLK 512
#define NTHR 256
#define TOTAL (NBLK * NTHR)
#define TOL_F 1e-4f
#define MAX_ITER 1000
#define MIN_REL_DIFF 1e-5f

static_assert(NBLK == 2 * NTHR, "block-0 final reduce assumes NBLK == 2*NTHR");

typedef float v2f __attribute__((ext_vector_type(2)));
typedef float v8f __attribute__((ext_vector_type(8)));

struct Ctrl {
  unsigned bar_count;   // grid barrier arrival counter
  unsigned bar_gen;     // grid barrier generation
  float lo, hi, mid;
  int active;
  float vmax, vmin;
  float sum_v;
  int degenerate;
  int too_small;
  int pad;
  double size;          // 8-byte aligned (offset 48)
};

// ---- device-scope (L2-coherent, vector-path) scalar accessors -------------
template <typename T>
__device__ __forceinline__ T ld_dev(T* p) {
  return __hip_atomic_load(p, __ATOMIC_RELAXED, __HIP_MEMORY_SCOPE_AGENT);
}
template <typename T>
__device__ __forceinline__ void st_dev(T* p, T v) {
  __hip_atomic_store(p, v, __ATOMIC_RELAXED, __HIP_MEMORY_SCOPE_AGENT);
}

// ---- Kahan compensated accumulation ---------------------------------------
__device__ __forceinline__ void kadd(float& s, float& c, float x) {
  float y = x - c;
  float t = s + y;
  c = (t - s) - y;
  s = t;
}

// ---- 32-lane sum via one WMMA: D = A(16x4) * ones(4x16) -------------------
// Each lane contributes its value once into A; with B == all-ones,
// sum over any full D column == sum over all A elements == wave sum.
// Lanes 0-15 hold rows M=0..7 of a column, lanes 16-31 rows M=8..15,
// so t + shfl_xor(t,16) gives the total on every lane.
__device__ __forceinline__ float wave_sum32(float x) {
  v2f a; a[0] = x;    a[1] = 0.0f;
  v2f b; b[0] = 1.0f; b[1] = 1.0f;
  v8f c = {0.f, 0.f, 0.f, 0.f, 0.f, 0.f, 0.f, 0.f};
  v8f d = __builtin_amdgcn_wmma_f32_16x16x4_f32(false, a, false, b,
                                                (short)0, c, false, false);
  float t = ((d[0] + d[1]) + (d[2] + d[3])) + ((d[4] + d[5]) + (d[6] + d[7]));
  t += __shfl_xor(t, 16, 32);
  return t;
}

__device__ __forceinline__ float block_sum(float x, float* sred) {
  float w = wave_sum32(x);
  __syncthreads();
  if ((threadIdx.x & 31) == 0) sred[threadIdx.x >> 5] = w;
  __syncthreads();
  float r = 0.0f;
  if (threadIdx.x == 0) {
#pragma unroll
    for (int i = 0; i < NTHR / 32; ++i) r += sred[i];
  }
  return r;  // valid in thread 0 only
}

__device__ __forceinline__ float block_max(float x, float* sred) {
#pragma unroll
  for (int o = 16; o > 0; o >>= 1) x = fmaxf(x, __shfl_xor(x, o, 32));
  __syncthreads();
  if ((threadIdx.x & 31) == 0) sred[threadIdx.x >> 5] = x;
  __syncthreads();
  float r = -3.402823466e38f;
  if (threadIdx.x == 0) {
#pragma unroll
    for (int i = 0; i < NTHR / 32; ++i) r = fmaxf(r, sred[i]);
  }
  return r;
}

__device__ __forceinline__ float block_min(float x, float* sred) {
#pragma unroll
  for (int o = 16; o > 0; o >>= 1) x = fminf(x, __shfl_xor(x, o, 32));
  __syncthreads();
  if ((threadIdx.x & 31) == 0) sred[threadIdx.x >> 5] = x;
  __syncthreads();
  float r = 3.402823466e38f;
  if (threadIdx.x == 0) {
#pragma unroll
    for (int i = 0; i < NTHR / 32; ++i) r = fminf(r, sred[i]);
  }
  return r;
}

// ---- grid-wide barrier (init kernel zeroes counters each launch) ----------
__device__ __forceinline__ void grid_sync(Ctrl* c, unsigned* gen_local) {
  __syncthreads();
  if (threadIdx.x == 0) {
    unsigned target = ++(*gen_local);
    unsigned prev = __hip_atomic_fetch_add(&c->bar_count, 1u, __ATOMIC_ACQ_REL,
                                           __HIP_MEMORY_SCOPE_AGENT);
    if (prev == (unsigned)(NBLK - 1)) {
      __hip_atomic_store(&c->bar_count, 0u, __ATOMIC_RELAXED,
                         __HIP_MEMORY_SCOPE_AGENT);
      __hip_atomic_fetch_add(&c->bar_gen, 1u, __ATOMIC_ACQ_REL,
                             __HIP_MEMORY_SCOPE_AGENT);
    } else {
      while (__hip_atomic_load(&c->bar_gen, __ATOMIC_ACQUIRE,
                               __HIP_MEMORY_SCOPE_AGENT) < target) {
        __builtin_amdgcn_s_sleep(2);
      }
    }
  }
  __syncthreads();
}

// ---- streaming pass: MODE 0 = {sum,sumsq,max,min}; 1 = {pp, pp^2};
//      2 = {pp, pp*v} ------------------------------------------------------
template <int MODE>
__device__ __forceinline__ void accum_elem(float xv, float eta, float& s1,
                                           float& c1, float& s2, float& c2,
                                           float& mx, float& mn) {
  if (MODE == 0) {
    mx = fmaxf(mx, xv);
    mn = fminf(mn, xv);
    kadd(s1, c1, xv);
    kadd(s2, c2, xv * xv);
  } else if (MODE == 1) {
    float pp = fmaxf(xv - eta, 0.0f);
    kadd(s1, c1, pp);
    kadd(s2, c2, pp * pp);
  } else {
    float pp = fmaxf(xv - eta, 0.0f);
    kadd(s1, c1, pp);
    kadd(s2, c2, pp * xv);
  }
}

template <int MODE>
__device__ __forceinline__ void pass_accum(const float4* __restrict__ v4,
                                           int nv4, int gtid, float eta,
                                           float& r1, float& r2, float& rmx,
                                           float& rmn) {
  float s1 = 0.f, c1 = 0.f, s2 = 0.f, c2 = 0.f;
  float mx = -3.402823466e38f, mn = 3.402823466e38f;
  int idx = gtid;
  // unrolled-by-4 grid-stride: 4 outstanding b128 loads + light prefetch
  for (; idx + 3 * TOTAL < nv4; idx += 4 * TOTAL) {
    int pf = idx + 8 * TOTAL;
    if (pf < nv4) __builtin_prefetch((const void*)(v4 + pf), 0, 1);
    float4 x0 = v4[idx];
    float4 x1 = v4[idx + TOTAL];
    float4 x2 = v4[idx + 2 * TOTAL];
    float4 x3 = v4[idx + 3 * TOTAL];
    float ev[16] = {x0.x, x0.y, x0.z, x0.w, x1.x, x1.y, x1.z, x1.w,
                    x2.x, x2.y, x2.z, x2.w, x3.x, x3.y, x3.z, x3.w};
#pragma unroll
    for (int j = 0; j < 16; ++j)
      accum_elem<MODE>(ev[j], eta, s1, c1, s2, c2, mx, mn);
  }
  for (; idx < nv4; idx += TOTAL) {
    float4 x = v4[idx];
    float ev[4] = {x.x, x.y, x.z, x.w};
#pragma unroll
    for (int j = 0; j < 4; ++j)
      accum_elem<MODE>(ev[j], eta, s1, c1, s2, c2, mx, mn);
  }
  r1 = s1 + c1;
  r2 = s2 + c2;
  rmx = mx;
  rmn = mn;
}

// ---------------------------------------------------------------------------
__global__ void dvp_init(Ctrl* c) {
  c->bar_count = 0u;
  c->bar_gen = 0u;
}

__global__ __launch_bounds__(NTHR, 1) void dvp_main(
    const float* __restrict__ v, float* __restrict__ out,
    char* __restrict__ ws, int n) {
  Ctrl* ctrl = (Ctrl*)ws;
  float* pA = (float*)(ws + 256);
  float* pB = pA + NBLK;
  float* pC = pB + NBLK;
  float* pD = pC + NBLK;
  const float4* v4 = (const float4*)v;
  const int nv4 = n >> 2;
  const int tid = threadIdx.x;
  const int bid = blockIdx.x;
  const int gtid = bid * NTHR + tid;
  __shared__ float sred[NTHR / 32];
  unsigned genc = 0;

  // -------- phase 0: sum, sumsq, max, min --------
  {
    float s1, s2, mx, mn;
    pass_accum<0>(v4, nv4, gtid, 0.0f, s1, s2, mx, mn);
    float bs1 = block_sum(s1, sred);
    float bs2 = block_sum(s2, sred);
    float bmx = block_max(mx, sred);
    float bmn = block_min(mn, sred);
    if (tid == 0) {
      pA[bid] = bs1;
      pB[bid] = bs2;
      pC[bid] = bmx;
      pD[bid] = bmn;
    }
  }
  grid_sync(ctrl, &genc);
  if (bid == 0) {
    float a = pA[tid] + pA[tid + NTHR];
    float b = pB[tid] + pB[tid + NTHR];
    float cmx = fmaxf(pC[tid], pC[tid + NTHR]);
    float cmn = fminf(pD[tid], pD[tid + NTHR]);
    float S = block_sum(a, sred);
    float SS = block_sum(b, sred);
    float VX = block_max(cmx, sred);
    float VN = block_min(cmn, sred);
    if (tid == 0) {
      double m = (double)n;
      double var = ((double)SS - (double)S * (double)S / m) / (m - 1.0);
      double size = var;  // LMBDA^2 * var^(2*GAMMA-1) with LMBDA=GAMMA=1
      float lo = (float)(-(1.0 / (sqrt(2.0 * size + 1.0) - 1.0)) * (double)VX);
      float hi = VX;
      int degenerate = ((VX - VN) / VX <= MIN_REL_DIFF) ? 1 : 0;
      int too_small = (m <= 1.0 + 2.0 * size) ? 1 : 0;
      int active =
          (!degenerate && !too_small && (hi - lo > TOL_F)) ? 1 : 0;
      st_dev(&ctrl->lo, lo);
      st_dev(&ctrl->hi, hi);
      st_dev(&ctrl->mid, 0.5f * (lo + hi));
      st_dev(&ctrl->active, active);
      st_dev(&ctrl->vmax, VX);
      st_dev(&ctrl->vmin, VN);
      st_dev(&ctrl->sum_v, S);
      st_dev(&ctrl->degenerate, degenerate);
      st_dev(&ctrl->too_small, too_small);
      st_dev(&ctrl->size, size);
    }
  }
  grid_sync(ctrl, &genc);

  // -------- bisection: while (hi-lo > TOL && i < MAX_ITER) --------
  for (int it = 0; it < MAX_ITER; ++it) {
    if (ld_dev(&ctrl->active) == 0) break;
    float eta = ld_dev(&ctrl->mid);
    float t1, t2, dmx, dmn;
    pass_accum<1>(v4, nv4, gtid, eta, t1, t2, dmx, dmn);
    float b1 = block_sum(t1, sred);
    float b2 = block_sum(t2, sred);
    if (tid == 0) {
      pA[bid] = b1;
      pB[bid] = b2;
    }
    grid_sync(ctrl, &genc);
    if (bid == 0) {
      float a = pA[tid] + pA[tid + NTHR];
      float b = pB[tid] + pB[tid + NTHR];
      float S1 = block_sum(a, sred);
      float S2 = block_sum(b, sred);
      if (tid == 0) {
        double m = (double)n;
        // mean((m*p-1)^2) - size  ==  m*S2/S1^2 - 1 - size
        double target =
            m * (double)S2 / ((double)S1 * (double)S1) - 1.0 - ld_dev(&ctrl->size);
        float lo = ld_dev(&ctrl->lo);
        float hi = ld_dev(&ctrl->hi);
        float mid = ld_dev(&ctrl->mid);
        if (target > 0.0) hi = mid; else lo = mid;
        mid = 0.5f * (lo + hi);
        st_dev(&ctrl->lo, lo);
        st_dev(&ctrl->hi, hi);
        st_dev(&ctrl->mid, mid);
        st_dev(&ctrl->active,
               ((hi - lo > TOL_F) && (it + 1 < MAX_ITER)) ? 1 : 0);
      }
    }
    grid_sync(ctrl, &genc);
  }

  // -------- final: q = p(0.5*(lo+hi)); out = dot(q, v) --------
  {
    float eta = ld_dev(&ctrl->mid);  // mid is maintained as 0.5*(lo+hi)
    float f1, f2, dmx, dmn;
    pass_accum<2>(v4, nv4, gtid, eta, f1, f2, dmx, dmn);
    float b1 = block_sum(f1, sred);
    float b2 = block_sum(f2, sred);
    if (tid == 0) {
      pA[bid] = b1;
      pB[bid] = b2;
    }
  }
  grid_sync(ctrl, &genc);
  if (bid == 0) {
    float a = pA[tid] + pA[tid + NTHR];
    float b = pB[tid] + pB[tid + NTHR];
    float SP = block_sum(a, sred);
    float SPV = block_sum(b, sred);
    if (tid == 0) {
      float r = SPV / SP;                          // bisection branch
      if (ld_dev(&ctrl->too_small)) r = ld_dev(&ctrl->vmax);   // argmax: == vmax
      if (ld_dev(&ctrl->degenerate)) r = ld_dev(&ctrl->sum_v) / (float)n;
      out[0] = r;
    }
  }
}

// ---------------------------------------------------------------------------
extern "C" void kernel_launch(void* const* d_in, const int* in_sizes, int n_in,
                              void* d_out, int out_size, void* d_ws,
                              size_t ws_size, hipStream_t stream) {
  const float* v = (const float*)d_in[0];
  float* out = (float*)d_out;
  int n = in_sizes[0];
  (void)n_in; (void)out_size; (void)ws_size;
  Ctrl* ctrl = (Ctrl*)d_ws;
  dvp_init<<<1, 1, 0, stream>>>(ctrl);
  dvp_main<<<NBLK, NTHR, 0, stream>>>(v, out, (char*)d_ws, n);
}